// GATv2Conv_56908316672629
// MI455X (gfx1250) — compile-verified
//
#include <hip/hip_runtime.h>

// ---------------------------------------------------------------------------
// GATv2Conv for MI455X (gfx1250, wave32).
//   N=100000 nodes, D_IN=128, H=4 heads, C=32 ch, E=1.6M edges.
// Pipeline:
//   k_init      : out = bias broadcast, m = -inf, denom = 0
//   k_gemm      : xl = x@W_l, xr = x@W_r   (V_WMMA_F32_16X16X4_F32)
//   k_edge_score: score[e,h] = att_h . leakyrelu(xl[src]+xr[dst]); segmax -> m
//   k_edge_exp  : ex = exp(score - m[dst]); segsum -> denom
//   k_edge_agg  : out[dst] += (ex/denom) * xl[src]   (f32 atomics, L2-resident)
// ---------------------------------------------------------------------------

typedef __attribute__((ext_vector_type(2))) float v2f;
typedef __attribute__((ext_vector_type(8))) float v8f;

#define NEG_SLOPE 0.2f

// ---------------- init: out = bias, m = -inf, denom = 0 --------------------
__global__ __launch_bounds__(256) void k_init(float* __restrict__ out,
                                              unsigned* __restrict__ mmax_bits,
                                              float* __restrict__ denom,
                                              const float* __restrict__ bias,
                                              long long N) {
  long long i = (long long)blockIdx.x * 256 + threadIdx.x;
  if (i < N * 128) out[i] = bias[i & 127];
  if (i < N * 4) {
    mmax_bits[i] = 0xFF800000u;  // -inf
    denom[i] = 0.0f;
  }
}

// ---------------- dual GEMM: [16 rows] x [128 cols], f32 WMMA --------------
// blockIdx.y: 0 -> (W_l, xl), 1 -> (W_r, xr). 8 waves, one 16-col tile each.
__global__ __launch_bounds__(256) void k_gemm(const float* __restrict__ x,
                                              const float* __restrict__ Wl,
                                              const float* __restrict__ Wr,
                                              float* __restrict__ xl,
                                              float* __restrict__ xr, int N) {
  const float* __restrict__ W = (blockIdx.y == 0) ? Wl : Wr;
  float* __restrict__ out = (blockIdx.y == 0) ? xl : xr;

  __shared__ float sA[16][132];  // 16x128 x-tile, +4 pad => conflict-free

  const int tid = threadIdx.x;
  const int r0 = blockIdx.x * 16;

  // cooperative stage of the 16x128 A tile into LDS
  for (int i = tid; i < 16 * 128; i += 256) {
    int r = i >> 7, c = i & 127;
    sA[r][c] = (r0 + r < N) ? x[(size_t)(r0 + r) * 128 + c] : 0.0f;
  }
  __syncthreads();

  const int wave = tid >> 5;       // 0..7 -> col tile
  const int lane = tid & 31;
  const int hl = lane >> 4;        // half-wave selector (K pairs)
  const int ln = lane & 15;
  const int c0 = wave * 16;

  v8f acc = {};
#pragma unroll
  for (int k = 0; k < 128; k += 4) {
    // A 16x4 frag: lanes 0-15 -> K=k,k+1 ; lanes 16-31 -> K=k+2,k+3 (M=ln)
    v2f a;
    a.x = sA[ln][k + 2 * hl];
    a.y = sA[ln][k + 2 * hl + 1];
    // B 4x16 frag (mirror layout), read from L2-hot 64KB weight matrix
    const float* wp = W + (size_t)(k + 2 * hl) * 128 + c0 + ln;
    v2f b;
    b.x = wp[0];
    b.y = wp[128];
    acc = __builtin_amdgcn_wmma_f32_16x16x4_f32(false, a, false, b, (short)0,
                                                acc, false, false);
  }

  // C/D layout: VGPR i -> row i (lanes 0-15) / row i+8 (lanes 16-31)
#pragma unroll
  for (int i = 0; i < 8; ++i) {
    int row = r0 + i + 8 * hl;
    if (row < N) out[(size_t)row * 128 + c0 + ln] = acc[i];
  }
}

// monotonic float atomic-max via signed-max / unsigned-min bit trick
__device__ __forceinline__ void atomicMaxFloat(float* addr, float val) {
  if (val >= 0.0f)
    atomicMax((int*)addr, __float_as_int(val));
  else
    atomicMin((unsigned int*)addr, __float_as_uint(val));
}

// ------------- pass 1: edge scores + segment max (wave per edge) -----------
__global__ __launch_bounds__(256) void k_edge_score(
    const long long* __restrict__ ei, const float* __restrict__ xl,
    const float* __restrict__ xr, const float* __restrict__ att,
    float* __restrict__ score, float* __restrict__ mmax, long long E) {
  const int wid = threadIdx.x >> 5, lane = threadIdx.x & 31;
  const long long e = (long long)blockIdx.x * 8 + wid;
  if (e >= E) return;
  const long long s = ei[e], d = ei[E + e];
  const int ch = lane * 4;         // each lane: 4 channels
  const int head = lane >> 3;      // 8 lanes per head (32 ch)

  const float4 xj = *(const float4*)(xl + (size_t)s * 128 + ch);
  const float4 xi = *(const float4*)(xr + (size_t)d * 128 + ch);
  float4 v;
  v.x = xi.x + xj.x; v.y = xi.y + xj.y;
  v.z = xi.z + xj.z; v.w = xi.w + xj.w;
  v.x = v.x > 0.f ? v.x : NEG_SLOPE * v.x;
  v.y = v.y > 0.f ? v.y : NEG_SLOPE * v.y;
  v.z = v.z > 0.f ? v.z : NEG_SLOPE * v.z;
  v.w = v.w > 0.f ? v.w : NEG_SLOPE * v.w;

  const float4 a = *(const float4*)(att + head * 32 + (lane & 7) * 4);
  float p = v.x * a.x + v.y * a.y + v.z * a.z + v.w * a.w;
  // reduce across the 8 lanes of this head (stays inside the group)
  p += __shfl_xor(p, 4, 32);
  p += __shfl_xor(p, 2, 32);
  p += __shfl_xor(p, 1, 32);

  if ((lane & 7) == 0) {
    score[e * 4 + head] = p;
    atomicMaxFloat(mmax + (size_t)d * 4 + head, p);
  }
}

// ------------- pass 2: exp + segment sum (thread per edge*head) ------------
__global__ __launch_bounds__(256) void k_edge_exp(const long long* __restrict__ ei,
                                                  const float* __restrict__ mmax,
                                                  float* __restrict__ score,
                                                  float* __restrict__ denom,
                                                  long long E) {
  long long id = (long long)blockIdx.x * 256 + threadIdx.x;
  if (id >= E * 4) return;
  long long e = id >> 2;
  int h = (int)(id & 3);
  long long d = ei[E + e];
  float ex = __expf(score[id] - mmax[(size_t)d * 4 + h]);
  score[id] = ex;  // overwrite score with exp value
  atomicAdd(denom + (size_t)d * 4 + h, ex);
}

// ------------- pass 3: alpha-weighted scatter add (wave per edge) ----------
__global__ __launch_bounds__(256) void k_edge_agg(const long long* __restrict__ ei,
                                                  const float* __restrict__ xl,
                                                  const float* __restrict__ score,
                                                  const float* __restrict__ denom,
                                                  float* __restrict__ out,
                                                  long long E) {
  const int wid = threadIdx.x >> 5, lane = threadIdx.x & 31;
  const long long e = (long long)blockIdx.x * 8 + wid;
  if (e >= E) return;
  const long long s = ei[e], d = ei[E + e];
  const int ch = lane * 4;
  const int head = lane >> 3;

  const float alpha =
      score[e * 4 + head] / (denom[(size_t)d * 4 + head] + 1e-16f);
  const float4 xj = *(const float4*)(xl + (size_t)s * 128 + ch);
  float* o = out + (size_t)d * 128 + ch;
  atomicAdd(o + 0, alpha * xj.x);
  atomicAdd(o + 1, alpha * xj.y);
  atomicAdd(o + 2, alpha * xj.z);
  atomicAdd(o + 3, alpha * xj.w);
}

// ---------------------------------------------------------------------------
extern "C" void kernel_launch(void* const* d_in, const int* in_sizes, int n_in,
                              void* d_out, int out_size, void* d_ws,
                              size_t ws_size, hipStream_t stream) {
  const float* x = (const float*)d_in[0];
  const long long* ei = (const long long*)d_in[1];  // [2,E] int64: src row, dst row
  const float* Wl = (const float*)d_in[2];
  const float* Wr = (const float*)d_in[3];
  const float* att = (const float*)d_in[4];
  const float* bias = (const float*)d_in[5];
  float* out = (float*)d_out;

  const long long N = in_sizes[0] / 128;
  const long long E = in_sizes[1] / 2;

  // workspace layout (≈131 MB): xl | xr | score/ex | m | denom
  float* xl = (float*)d_ws;
  float* xr = xl + (size_t)N * 128;
  float* score = xr + (size_t)N * 128;
  float* mmax = score + (size_t)E * 4;
  float* denom = mmax + (size_t)N * 4;

  dim3 ginit((unsigned)((N * 128 + 255) / 256));
  k_init<<<ginit, 256, 0, stream>>>(out, (unsigned*)mmax, denom, bias, N);

  dim3 ggemm((unsigned)((N + 15) / 16), 2);
  k_gemm<<<ggemm, 256, 0, stream>>>(x, Wl, Wr, xl, xr, (int)N);

  unsigned gedge = (unsigned)((E + 7) / 8);
  k_edge_score<<<gedge, 256, 0, stream>>>(ei, xl, xr, att, score, mmax, E);

  unsigned gexp = (unsigned)((E * 4 + 255) / 256);
  k_edge_exp<<<gexp, 256, 0, stream>>>(ei, mmax, score, denom, E);

  k_edge_agg<<<gedge, 256, 0, stream>>>(ei, xl, score, denom, out, E);
}